// FraudDetectionHybridModel_65481071400680
// MI455X (gfx1250) — compile-verified
//
#include <hip/hip_runtime.h>
#include <math.h>

// ---------------------------------------------------------------------------
// FraudDetectionHybridModel for MI455X (gfx1250, wave32)
//
// dist^2 = ||phi(a)||^2 + ||phi(b)||^2 - 2 a.b  with  ||phi(.)||^2 == 2 exactly
// => k = exp(2*(a.b) - 4) = exp2(dot*2log2e - 4log2e).  The N x M x K=4 GEMM
// maps 1:1 onto V_WMMA_F32_16X16X4_F32 (A 16x4 f32, B 4x16 f32, C/D 16x16 f32).
// Hot loop is v_exp_f32-throughput bound (67M exps); all data lives in L2.
// exp2 constants are folded into the pk_fma so each kernel eval is exactly
// one fma + one v_exp_f32.
// ---------------------------------------------------------------------------

typedef __attribute__((ext_vector_type(2))) float v2f;
typedef __attribute__((ext_vector_type(8))) float v8f;

#define N_     16384
#define M_     4096
#define L_     8
#define CHUNKS 8      // split of M per 16-row strip (parallelism + atomics)
#define TILE   16

// exp(2*d - 4) == exp2(d * C1 + C0)
#define LOG2E_F 1.4426950408889634f
#define C1_     (2.0f * LOG2E_F)   //  2.8853900817779268
#define C0_     (-4.0f * LOG2E_F)  // -5.7707801635558535

// --- Kernel 1: chained tiny layers + feature map; also zero the row sums ---
__global__ void feats_kernel(const float* __restrict__ x,   // [N,2]
                             const float* __restrict__ W,   // [L,2,2]
                             const float* __restrict__ b,   // [L,2]
                             const float* __restrict__ s,   // [L,2]
                             const float* __restrict__ t,   // [L,2]
                             float* __restrict__ a,         // [N,4] out
                             float* __restrict__ rowsum)    // [N]   zeroed
{
    int n = blockIdx.x * blockDim.x + threadIdx.x;
    if (n >= N_) return;
    float h0 = x[2 * n + 0];
    float h1 = x[2 * n + 1];
#pragma unroll
    for (int l = 0; l < L_; ++l) {
        const float w00 = W[4 * l + 0], w01 = W[4 * l + 1];
        const float w10 = W[4 * l + 2], w11 = W[4 * l + 3];
        float u0 = fmaf(w00, h0, fmaf(w01, h1, b[2 * l + 0]));
        float u1 = fmaf(w10, h0, fmaf(w11, h1, b[2 * l + 1]));
        h0 = fmaf(tanhf(u0), s[2 * l + 0], t[2 * l + 0]);
        h1 = fmaf(tanhf(u1), s[2 * l + 1], t[2 * l + 1]);
    }
    // phi(z) = [sin z0, sin z1, cos z0, cos z1]
    a[4 * n + 0] = sinf(h0);
    a[4 * n + 1] = sinf(h1);
    a[4 * n + 2] = cosf(h0);
    a[4 * n + 3] = cosf(h1);
    rowsum[n] = 0.0f;
}

// --- Kernel 2: feature map of the reference set -----------------------------
__global__ void bref_kernel(const float* __restrict__ ref,  // [M,2]
                            float* __restrict__ bref)       // [M,4] out
{
    int m = blockIdx.x * blockDim.x + threadIdx.x;
    if (m >= M_) return;
    float r0 = ref[2 * m + 0];
    float r1 = ref[2 * m + 1];
    bref[4 * m + 0] = sinf(r0);
    bref[4 * m + 1] = sinf(r1);
    bref[4 * m + 2] = cosf(r0);
    bref[4 * m + 3] = cosf(r1);
}

// --- Kernel 3: WMMA 16x16x4 f32 GEMM tiles + exp + row-sum reduction --------
// One wave = one 16-row strip of N and a (M_/CHUNKS)-wide chunk of M.
// A-frag (16x4 f32): lanes 0-15 hold row m0+l K={0,1}; lanes 16-31 K={2,3}.
// B-frag (4x16 f32) = bref tile transposed: same lane striping over bref rows.
// D (16x16 f32): vgpr r = row r (lanes 0-15) / row r+8 (lanes 16-31).
__global__ void kexp_kernel(const float* __restrict__ a,    // [N,4]
                            const float* __restrict__ bref, // [M,4]
                            float* __restrict__ rowsum)     // [N] atomic acc
{
    const int lane = threadIdx.x & 31;
    const int wave = (blockIdx.x * blockDim.x + threadIdx.x) >> 5;
    const int strip = wave / CHUNKS;            // 16-row strip index in N
    const int chunk = wave % CHUNKS;            // chunk index in M
    const int m0 = strip * TILE;
    const int mchunk = M_ / CHUNKS;             // 512 columns per wave
    const int c_begin = chunk * mchunk;
    const int c_end = c_begin + mchunk;

    const int half = lane >> 4;                 // 0: K=0,1   1: K=2,3
    const int l16 = lane & 15;

    // A fragment is invariant across the whole M loop for this wave.
    v2f afrag;
    {
        const float* ap = a + (size_t)(m0 + l16) * 4 + half * 2;
        afrag.x = ap[0];
        afrag.y = ap[1];
    }

    float acc[8];
#pragma unroll
    for (int r = 0; r < 8; ++r) acc[r] = 0.0f;

    for (int c0 = c_begin; c0 < c_end; c0 += TILE) {
        v2f bfrag;
        const float* bp = bref + (size_t)(c0 + l16) * 4 + half * 2;
        bfrag.x = bp[0];
        bfrag.y = bp[1];

        v8f d = {};
        // (neg_a, A, neg_b, B, c_mod, C, reuse_a, reuse_b)
        d = __builtin_amdgcn_wmma_f32_16x16x4_f32(
                false, afrag, false, bfrag, (short)0, d, false, false);

#pragma unroll
        for (int r = 0; r < 8; ++r) {
            // k = exp(2*dot - 4) = exp2(dot*C1 + C0): one fma + one v_exp_f32
            acc[r] += __builtin_amdgcn_exp2f(fmaf(d[r], C1_, C0_));
        }
    }

    // Sum each row over the 16 column-lanes of its half-wave
    // (xor masks 1,2,4,8 stay within lanes [0,16) and [16,32)).
#pragma unroll
    for (int r = 0; r < 8; ++r) {
        float v = acc[r];
        v += __shfl_xor(v, 1, 32);
        v += __shfl_xor(v, 2, 32);
        v += __shfl_xor(v, 4, 32);
        v += __shfl_xor(v, 8, 32);
        acc[r] = v;
    }
    if (l16 == 0) {
        const int rowbase = m0 + half * 8;      // lane0 -> rows 0-7, lane16 -> rows 8-15
#pragma unroll
        for (int r = 0; r < 8; ++r)
            atomicAdd(&rowsum[rowbase + r], acc[r]);
    }
}

// --- Kernel 4: mean, tiny MLP head, stable softmax --------------------------
__global__ void head_kernel(const float* __restrict__ rowsum, // [N]
                            const float* __restrict__ W1,     // [4,1]
                            const float* __restrict__ b1,     // [4]
                            const float* __restrict__ W2,     // [2,4]
                            const float* __restrict__ b2,     // [2]
                            float* __restrict__ out)          // [N,2]
{
    int n = blockIdx.x * blockDim.x + threadIdx.x;
    if (n >= N_) return;
    float agg = rowsum[n] * (1.0f / (float)M_);
    float h[4];
#pragma unroll
    for (int i = 0; i < 4; ++i)
        h[i] = tanhf(fmaf(W1[i], agg, b1[i]));
    float lg0 = b2[0], lg1 = b2[1];
#pragma unroll
    for (int i = 0; i < 4; ++i) {
        lg0 = fmaf(W2[i],     h[i], lg0);
        lg1 = fmaf(W2[4 + i], h[i], lg1);
    }
    float mx = fmaxf(lg0, lg1);
    float e0 = __builtin_amdgcn_exp2f((lg0 - mx) * LOG2E_F);
    float e1 = __builtin_amdgcn_exp2f((lg1 - mx) * LOG2E_F);
    float inv = 1.0f / (e0 + e1);
    out[2 * n + 0] = e0 * inv;
    out[2 * n + 1] = e1 * inv;
}

// ---------------------------------------------------------------------------
extern "C" void kernel_launch(void* const* d_in, const int* in_sizes, int n_in,
                              void* d_out, int out_size, void* d_ws, size_t ws_size,
                              hipStream_t stream) {
    const float* x     = (const float*)d_in[0];  // [N,2]
    const float* lW    = (const float*)d_in[1];  // [L,2,2]
    const float* lb    = (const float*)d_in[2];  // [L,2]
    const float* ls    = (const float*)d_in[3];  // [L,2]
    const float* lt    = (const float*)d_in[4];  // [L,2]
    const float* ref   = (const float*)d_in[5];  // [M,2]
    const float* W1    = (const float*)d_in[6];  // [4,1]
    const float* b1    = (const float*)d_in[7];  // [4]
    const float* W2    = (const float*)d_in[8];  // [2,4]
    const float* b2    = (const float*)d_in[9];  // [2]
    float*       out   = (float*)d_out;          // [N,2]

    // Workspace layout (floats): a[N*4] | rowsum[N] | bref[M*4]
    float* a      = (float*)d_ws;
    float* rowsum = a + (size_t)N_ * 4;
    float* bref   = rowsum + (size_t)N_;

    feats_kernel<<<N_ / 256, 256, 0, stream>>>(x, lW, lb, ls, lt, a, rowsum);
    bref_kernel<<<M_ / 256, 256, 0, stream>>>(ref, bref);

    // waves = (N/16) strips * CHUNKS = 8192 waves = 262144 threads
    const int waves = (N_ / TILE) * CHUNKS;
    const int threads = waves * 32;
    kexp_kernel<<<threads / 256, 256, 0, stream>>>(a, bref, rowsum);

    head_kernel<<<N_ / 256, 256, 0, stream>>>(rowsum, W1, b1, W2, b2, out);
}